// DistanceAwareGNN_46574625358117
// MI455X (gfx1250) — compile-verified
//
#include <hip/hip_runtime.h>
#include <hip/hip_bf16.h>

typedef float v2f __attribute__((ext_vector_type(2)));
typedef float v8f __attribute__((ext_vector_type(8)));

// ---------------------------------------------------------------------------
// Utility kernels
// ---------------------------------------------------------------------------
__global__ void fill_f32(float* __restrict__ p, float v, int n) {
    int i = blockIdx.x * blockDim.x + threadIdx.x;
    if (i < n) p[i] = v;
}

__global__ void fill_f32x4(float4* __restrict__ p, float v, int n4) {
    int i = blockIdx.x * blockDim.x + threadIdx.x;
    if (i < n4) p[i] = make_float4(v, v, v, v);
}

// dinv[i] = rsqrt(deg[i])   (deg >= 1 always: self-loop weight 1)
__global__ void rsqrt_kernel(const float* __restrict__ deg, float* __restrict__ dinv, int n) {
    int i = blockIdx.x * blockDim.x + threadIdx.x;
    if (i < n) dinv[i] = rsqrtf(deg[i]);
}

// ---------------------------------------------------------------------------
// Edge-weight MLP: ew = sigmoid(relu(a*Wd1+bd1) @ Wd2 + bd2); also accumulate
// weighted in-degree (deg pre-initialized to 1.0 for the self-loop).
// ---------------------------------------------------------------------------
__global__ void edge_mlp_deg(const float* __restrict__ edge_attr,
                             const int* __restrict__ dst,
                             const float* __restrict__ Wd1, const float* __restrict__ bd1,
                             const float* __restrict__ Wd2, const float* __restrict__ bd2,
                             float* __restrict__ ew, float* __restrict__ deg, int E) {
    int e = blockIdx.x * blockDim.x + threadIdx.x;
    if (e >= E) return;
    float a = edge_attr[e];
    float s = bd2[0];
#pragma unroll
    for (int j = 0; j < 16; ++j) {
        float h = fmaf(a, Wd1[j], bd1[j]);
        h = fmaxf(h, 0.0f);
        s = fmaf(h, Wd2[j], s);
    }
    float w = 1.0f / (1.0f + __expf(-s));
    ew[e] = w;
    atomicAdd(&deg[dst[e]], w);
}

// ---------------------------------------------------------------------------
// Edge scatter: agg[dst] += (dinv[src]*ew*dinv[dst]) * h[src]
// 16 threads per edge, float4 per thread: 128-bit coalesced gathers, four
// 32-bit atomic f32 adds. h (25.6MB) and agg (25.6MB) are L2-resident.
// ---------------------------------------------------------------------------
__global__ void scatter_edges(const float4* __restrict__ h4,
                              const float* __restrict__ ew,
                              const float* __restrict__ dinv,
                              const int* __restrict__ src,
                              const int* __restrict__ dst,
                              float* __restrict__ agg, int E) {
    int g = blockIdx.x * blockDim.x + threadIdx.x;
    int e = g >> 4;
    int q = g & 15;
    if (e >= E) return;
    int s = src[e];
    int d = dst[e];
    float norm = dinv[s] * ew[e] * dinv[d];
    float4 v = h4[(long long)s * 16 + q];
    float* out = agg + (long long)d * 64 + q * 4;
    atomicAdd(out + 0, norm * v.x);
    atomicAdd(out + 1, norm * v.y);
    atomicAdd(out + 2, norm * v.z);
    atomicAdd(out + 3, norm * v.w);
}

// Self-loop contribution: agg[i] += h[i] * dinv[i]^2  (== h[i]/deg[i])
__global__ void self_loop_add(const float4* __restrict__ h4,
                              const float* __restrict__ dinv,
                              float4* __restrict__ agg4, int N) {
    int g = blockIdx.x * blockDim.x + threadIdx.x;
    int i = g >> 4;
    int q = g & 15;
    if (i >= N) return;
    float di = dinv[i];
    float sc = di * di;
    float4 v = h4[(long long)i * 16 + q];
    float4 a = agg4[(long long)i * 16 + q];
    a.x += v.x * sc; a.y += v.y * sc; a.z += v.z * sc; a.w += v.w * sc;
    agg4[(long long)i * 16 + q] = a;
}

// In-place bias (+ optional ReLU): x = act(x + b[c]) over [N,64]
__global__ void bias_act(float4* __restrict__ x4, const float4* __restrict__ b4,
                         int relu, int N) {
    int g = blockIdx.x * blockDim.x + threadIdx.x;
    int i = g >> 4;
    int q = g & 15;
    if (i >= N) return;
    float4 v = x4[(long long)i * 16 + q];
    float4 b = b4[q];
    v.x += b.x; v.y += b.y; v.z += b.z; v.w += b.w;
    if (relu) {
        v.x = fmaxf(v.x, 0.0f); v.y = fmaxf(v.y, 0.0f);
        v.z = fmaxf(v.z, 0.0f); v.w = fmaxf(v.w, 0.0f);
    }
    x4[(long long)i * 16 + q] = v;
}

// ---------------------------------------------------------------------------
// f32 WMMA GEMM: C[rows x NCOL] = A[rows x 64] @ W[64 x NCOL] (+ bias)
// One wave per 16-row tile. K=64 -> 16 chained V_WMMA_F32_16X16X4_F32 per
// 16x16 output tile. NCOL is a template constant so all load offsets fold
// into immediate offsets on a single per-lane base pointer, and all 16 B
// fragments are hoisted ahead of the WMMA chain (single loadcnt wait).
// A-frag layout (32-bit A 16x4): lanes 0-15 hold K={k0,k0+1}, lanes 16-31
// hold K={k0+2,k0+3}, M = lane%16. B mirrored with N = lane%16.
// C/D: VGPR r -> M = r + (lane<16 ? 0 : 8), N = lane%16.
// EXEC stays all-ones (required for WMMA): loads clamped, stores guarded.
// ---------------------------------------------------------------------------
template <int NCOL>
__global__ __launch_bounds__(32) void gemm_wmma_f32(const float* __restrict__ A,
                                                    const float* __restrict__ W,
                                                    const float* __restrict__ bias,
                                                    float* __restrict__ C,
                                                    int rows) {
    const int lane = threadIdx.x & 31;
    const int row0 = blockIdx.x * 16;
    const int m    = lane & 15;
    const int kb   = (lane < 16) ? 0 : 2;   // K pair base within each 4-wide step
    const int mhi  = (lane < 16) ? 0 : 8;

    int arow = row0 + m;
    if (arow >= rows) arow = rows - 1;       // clamp: keep EXEC all-ones
    // A fragments: 8-byte loads at constant immediate offsets
    const v2f* ap2 = (const v2f*)(A + (long long)arow * 64 + kb);
    v2f afrag[16];
#pragma unroll
    for (int kk = 0; kk < 16; ++kk) afrag[kk] = ap2[kk * 2];

#pragma unroll
    for (int nt = 0; nt < (NCOL >> 4); ++nt) {
        const int col = nt * 16 + m;
        const float* Wp = W + kb * NCOL + col;   // per-lane base; rest immediate
        v2f bfrag[16];
#pragma unroll
        for (int kk = 0; kk < 16; ++kk) {
            bfrag[kk].x = Wp[(kk * 4 + 0) * NCOL];
            bfrag[kk].y = Wp[(kk * 4 + 1) * NCOL];
        }
        v8f acc = {};
#pragma unroll
        for (int kk = 0; kk < 16; ++kk) {
            acc = __builtin_amdgcn_wmma_f32_16x16x4_f32(
                false, afrag[kk], false, bfrag[kk], (short)0, acc, false, false);
        }
        float bv = (bias != nullptr) ? bias[col] : 0.0f;
        float* Cp = C + (long long)(row0 + mhi) * NCOL + col;
#pragma unroll
        for (int r = 0; r < 8; ++r) {
            if (row0 + mhi + r < rows) Cp[r * NCOL] = acc[r] + bv;
        }
    }
}

// ---------------------------------------------------------------------------
// Launch
// ---------------------------------------------------------------------------
extern "C" void kernel_launch(void* const* d_in, const int* in_sizes, int n_in,
                              void* d_out, int out_size, void* d_ws, size_t ws_size,
                              hipStream_t stream) {
    const float* x         = (const float*)d_in[0];   // [N,64]
    const int*   edge_idx  = (const int*)d_in[1];     // [2,E]
    const float* edge_attr = (const float*)d_in[2];   // [E,1]
    const float* Wd1       = (const float*)d_in[3];   // [1,16]
    const float* bd1       = (const float*)d_in[4];   // [16]
    const float* Wd2       = (const float*)d_in[5];   // [16,1]
    const float* bd2       = (const float*)d_in[6];   // [1]
    const float* W1c       = (const float*)d_in[7];   // [64,64]
    const float* b1c       = (const float*)d_in[8];   // [64]
    const float* W2c       = (const float*)d_in[9];   // [64,64]
    const float* b2c       = (const float*)d_in[10];  // [64]
    const float* Wo        = (const float*)d_in[11];  // [64,32]
    const float* bo        = (const float*)d_in[12];  // [32]

    const int N = in_sizes[0] / 64;
    const int E = in_sizes[2];
    const int* src = edge_idx;
    const int* dst = edge_idx + E;

    // workspace layout (floats)
    float* ws   = (float*)d_ws;
    float* ew   = ws;                        // E
    float* deg  = ew + E;                    // N
    float* dinv = deg + N;                   // N
    float* h    = dinv + N;                  // N*64 (GEMM outputs)
    float* agg  = h + (size_t)N * 64;        // N*64 (aggregation)

    const int T = 256;
    const int nodeBlocks  = (N + T - 1) / T;
    const int edgeBlocks  = (E + T - 1) / T;
    const int node16Blks  = (int)(((long long)N * 16 + T - 1) / T);  // float4 over [N,64]
    const int edge16Blks  = (int)(((long long)E * 16 + T - 1) / T);  // float4 over [E,64]
    const int zero4Blks   = (int)(((long long)N * 16 + T - 1) / T);
    const int gemmBlocks  = (N + 15) / 16;

    // deg = 1 (self-loop weight), agg = 0
    fill_f32<<<nodeBlocks, T, 0, stream>>>(deg, 1.0f, N);
    fill_f32x4<<<zero4Blks, T, 0, stream>>>((float4*)agg, 0.0f, N * 16);

    // edge weights + weighted degree
    edge_mlp_deg<<<edgeBlocks, T, 0, stream>>>(edge_attr, dst, Wd1, bd1, Wd2, bd2,
                                               ew, deg, E);
    rsqrt_kernel<<<nodeBlocks, T, 0, stream>>>(deg, dinv, N);

    // ---- conv1 ----
    gemm_wmma_f32<64><<<gemmBlocks, 32, 0, stream>>>(x, W1c, nullptr, h, N);
    scatter_edges<<<edge16Blks, T, 0, stream>>>((const float4*)h, ew, dinv, src, dst, agg, E);
    self_loop_add<<<node16Blks, T, 0, stream>>>((const float4*)h, dinv, (float4*)agg, N);
    bias_act<<<node16Blks, T, 0, stream>>>((float4*)agg, (const float4*)b1c, /*relu=*/1, N);

    // ---- conv2 ----  (h <- agg @ W2c, then agg re-zeroed and re-aggregated)
    gemm_wmma_f32<64><<<gemmBlocks, 32, 0, stream>>>(agg, W2c, nullptr, h, N);
    fill_f32x4<<<zero4Blks, T, 0, stream>>>((float4*)agg, 0.0f, N * 16);
    scatter_edges<<<edge16Blks, T, 0, stream>>>((const float4*)h, ew, dinv, src, dst, agg, E);
    self_loop_add<<<node16Blks, T, 0, stream>>>((const float4*)h, dinv, (float4*)agg, N);
    bias_act<<<node16Blks, T, 0, stream>>>((float4*)agg, (const float4*)b2c, /*relu=*/0, N);

    // ---- output head (fused bias) ----
    gemm_wmma_f32<32><<<gemmBlocks, 32, 0, stream>>>(agg, Wo, bo, (float*)d_out, N);
}